// Pi_72198400246426
// MI455X (gfx1250) — compile-verified
//
#include <hip/hip_runtime.h>
#include <hip/hip_bf16.h>

// out = inp + noise (elementwise, fp32). Memory-bound: ~402 MB @ 23.3 TB/s -> ~17us floor.
// B128 non-temporal loads/stores; exact-cover kernel has no bounds checks so the
// compiler emits a pure load-clause / add / store-clause stream (no exec masking,
// no VGPR zero-init chains).

typedef float v4f __attribute__((ext_vector_type(4)));

#define TPB 256
#define UNROLL 4

// Exact-cover: gridDim.x * TPB * UNROLL == n4 exactly. No guards.
__global__ __launch_bounds__(TPB) void Pi_add_noise_exact(const v4f* __restrict__ a,
                                                          const v4f* __restrict__ b,
                                                          v4f* __restrict__ o) {
    const int tid    = blockIdx.x * TPB + threadIdx.x;
    const int stride = gridDim.x * TPB;

    v4f x[UNROLL];
    v4f y[UNROLL];

    // 8 outstanding global_load_b128 (NT) before any use.
#pragma unroll
    for (int k = 0; k < UNROLL; ++k) {
        x[k] = __builtin_nontemporal_load(&a[tid + k * stride]);
        y[k] = __builtin_nontemporal_load(&b[tid + k * stride]);
    }

#pragma unroll
    for (int k = 0; k < UNROLL; ++k) {
        v4f r = x[k] + y[k];
        __builtin_nontemporal_store(r, &o[tid + k * stride]);
    }
}

// Guarded fallback for sizes that don't divide evenly.
__global__ __launch_bounds__(TPB) void Pi_add_noise_guarded(const v4f* __restrict__ a,
                                                            const v4f* __restrict__ b,
                                                            v4f* __restrict__ o,
                                                            int n4) {
    const int tid    = blockIdx.x * TPB + threadIdx.x;
    const int stride = gridDim.x * TPB;
#pragma unroll
    for (int k = 0; k < UNROLL; ++k) {
        const int i = tid + k * stride;
        if (i < n4) {
            v4f r = __builtin_nontemporal_load(&a[i]) + __builtin_nontemporal_load(&b[i]);
            __builtin_nontemporal_store(r, &o[i]);
        }
    }
}

// Scalar tail for n % 4 != 0 (not hit for 4096*8192, kept for generality).
__global__ void Pi_add_noise_tail(const float* __restrict__ a,
                                  const float* __restrict__ b,
                                  float* __restrict__ o,
                                  int base, int n) {
    int i = base + blockIdx.x * blockDim.x + threadIdx.x;
    if (i < n) {
        o[i] = a[i] + b[i];
    }
}

extern "C" void kernel_launch(void* const* d_in, const int* in_sizes, int n_in,
                              void* d_out, int out_size, void* d_ws, size_t ws_size,
                              hipStream_t stream) {
    const float* inp   = (const float*)d_in[0];
    const float* noise = (const float*)d_in[1];
    float* out         = (float*)d_out;

    const int n   = in_sizes[0];    // 4096 * 8192 = 33,554,432
    const int n4  = n >> 2;         // float4 count = 8,388,608
    const int rem = n & 3;

    const int per_block = TPB * UNROLL;   // 1024 float4 per block

    if (n4 > 0) {
        if (n4 % per_block == 0) {
            // Exact cover: 8192 blocks for the reference size. No guards in-kernel.
            Pi_add_noise_exact<<<n4 / per_block, TPB, 0, stream>>>(
                (const v4f*)inp, (const v4f*)noise, (v4f*)out);
        } else {
            Pi_add_noise_guarded<<<(n4 + per_block - 1) / per_block, TPB, 0, stream>>>(
                (const v4f*)inp, (const v4f*)noise, (v4f*)out, n4);
        }
    }
    if (rem > 0) {
        Pi_add_noise_tail<<<1, rem, 0, stream>>>(inp, noise, out, n4 << 2, n);
    }
}